// Model_43456479101401
// MI455X (gfx1250) — compile-verified
//
#include <hip/hip_runtime.h>
#include <math.h>

typedef float v2f __attribute__((ext_vector_type(2)));
typedef float v8f __attribute__((ext_vector_type(8)));

#define N_TOK        262144          // 16*128*128 tokens
#define KCODES       512
#define DIM          64
#define HW           16384           // 128*128
#define TOK_PER_BLK  128
#define NBLK1        (N_TOK / TOK_PER_BLK)   // 2048
#define EROW         68              // padded LDS row stride in floats (stride%64==4 -> conflict-free)

// d_out flat layout (all float32, reference return order)
#define OFF_LOSS 0
#define OFF_Q    1
#define OFF_PERP (1 + 16777216)      // 16777217
#define OFF_ENC  (OFF_PERP + 1)      // 16777218

// ---------------------------------------------------------------- init ----
__global__ void vq_init(int* __restrict__ counts) {
  counts[threadIdx.x] = 0;
}

// -------------------------------------------------------------- assign ----
// One block = 128 consecutive tokens (same batch image, contiguous HW).
// Wave w owns token rows [16w, 16w+16). Full K swept in 16-code tiles with
// chained V_WMMA_F32_16X16X4_F32 over D=64 (16 WMMAs per tile).
__global__ __launch_bounds__(256)
void vq_assign(const float* __restrict__ in, const float* __restrict__ emb,
               float* __restrict__ out_q, int* __restrict__ ws_idx,
               int* __restrict__ counts, float* __restrict__ ws_sse) {
  extern __shared__ float smem[];
  float* lds_e   = smem;                    // KCODES*EROW floats (padded codebook)
  float* lds_en  = smem + KCODES * EROW;    // KCODES  (||e||^2)
  float* red     = lds_en + KCODES;         // 256
  int*   lds_idx = (int*)(red + 256);       // TOK_PER_BLK

  const int tid  = threadIdx.x;
  const int lane = tid & 31;
  const int wave = tid >> 5;

  // stage codebook into padded LDS rows (coalesced global reads)
  for (int i = tid; i < KCODES * DIM; i += 256)
    lds_e[(i >> 6) * EROW + (i & 63)] = emb[i];
  __syncthreads();

  // per-code squared norms
  for (int r = tid; r < KCODES; r += 256) {
    float s = 0.f;
#pragma unroll
    for (int d2 = 0; d2 < DIM; ++d2) { float v = lds_e[r * EROW + d2]; s += v * v; }
    lds_en[r] = s;
  }
  __syncthreads();

  const int n0  = blockIdx.x * TOK_PER_BLK;     // first token of block
  const int b   = n0 >> 14;                     // batch index (HW = 2^14)
  const int hw0 = n0 & (HW - 1);                // hw offset (block never crosses b)
  const int tm  = wave * 16;                    // token-tile base in block
  const int m   = lane & 15;                    // row within tile / col within tile
  const int dl  = (lane >> 4) * 2;              // K-pair select for 16x16x4 frags

  // A fragments: x is NCHW, so dim stride is HW. lanes 0-15: K={0,1}, 16-31: K={2,3}
  const float* xb = in + ((size_t)b << 20) + (size_t)(hw0 + tm + m);
  v2f a[16];
#pragma unroll
  for (int c = 0; c < 16; ++c) {
    a[c].x = xb[(size_t)(4 * c + dl) * HW];
    a[c].y = xb[(size_t)(4 * c + dl + 1) * HW];
  }

  float bestVal[8];
  int   bestIdx[8];
#pragma unroll
  for (int r = 0; r < 8; ++r) { bestVal[r] = 3.4e38f; bestIdx[r] = 0; }

  for (int j = 0; j < KCODES / 16; ++j) {
    const int code = j * 16 + m;                // this lane's column code
    const float* eb = lds_e + code * EROW + dl; // B frag source (LDS, padded)
    v8f acc = {0.f, 0.f, 0.f, 0.f, 0.f, 0.f, 0.f, 0.f};
#pragma unroll
    for (int c = 0; c < 16; ++c) {
      v2f bf = *(const v2f*)(eb + 4 * c);       // ds_load_b64, 8B aligned
      acc = __builtin_amdgcn_wmma_f32_16x16x4_f32(false, a[c], false, bf,
                                                  (short)0, acc, false, false);
    }
    const float en = lds_en[code];
#pragma unroll
    for (int r = 0; r < 8; ++r) {
      // argmin target: ||e||^2 - 2 x.e  (||x||^2 constant per token)
      float dst = en - 2.0f * acc[r];
      if (dst < bestVal[r]) { bestVal[r] = dst; bestIdx[r] = code; }
    }
  }

  // butterfly min-reduce across the 16 lanes holding the 16 code residues
#pragma unroll
  for (int r = 0; r < 8; ++r) {
#pragma unroll
    for (int off = 1; off < 16; off <<= 1) {
      float ov = __shfl_xor(bestVal[r], off, 32);
      int   oi = __shfl_xor(bestIdx[r], off, 32);
      if (ov < bestVal[r] || (ov == bestVal[r] && oi < bestIdx[r])) {
        bestVal[r] = ov; bestIdx[r] = oi;
      }
    }
  }

  if ((lane & 15) == 0) {                       // lanes 0 and 16 publish 8 rows each
    const int half = lane >> 4;
#pragma unroll
    for (int r = 0; r < 8; ++r) {
      int trow = tm + r + 8 * half;
      int code = bestIdx[r];
      lds_idx[trow] = code;
      ws_idx[n0 + trow] = code;
      atomicAdd(&counts[code], 1);              // integer atomic: deterministic
    }
  }
  __syncthreads();

  // phase 2: scatter quantized (NCHW, coalesced along HW) + SSE partial
  float sse = 0.f;
  const float* inb = in + ((size_t)b << 20);
  float* qb = out_q + ((size_t)b << 20);
  for (int i = tid; i < TOK_PER_BLK * DIM; i += 256) {
    int d2 = i >> 7;                            // 0..63
    int t  = i & (TOK_PER_BLK - 1);
    int code = lds_idx[t];
    float val = lds_e[code * EROW + d2];
    size_t off = (size_t)d2 * HW + (size_t)(hw0 + t);
    float xv = inb[off];
    qb[off] = val;
    float df = val - xv;
    sse += df * df;
  }
  red[tid] = sse;
  __syncthreads();
  for (int s2 = 128; s2 > 0; s2 >>= 1) {
    if (tid < s2) red[tid] += red[tid + s2];
    __syncthreads();
  }
  if (tid == 0) ws_sse[blockIdx.x] = red[0];    // deterministic per-block partial
}

// -------------------------------------------------------------- encode ----
// One wave per one-hot row (512 floats). 537 MB of output -> non-temporal
// stores so we don't trash L2. Row base is only 8-mod-16 bytes (OFF_ENC is
// even-but-not-mult-of-4 floats), so use 8B float2 stores.
__global__ __launch_bounds__(256)
void vq_encode(const int* __restrict__ idx, float* __restrict__ enc) {
  const int lane = threadIdx.x & 31;
  const int wave = threadIdx.x >> 5;
  const size_t row = (size_t)blockIdx.x * 8 + wave;
  const int code = idx[row];
  float* p = enc + row * (size_t)KCODES;
#pragma unroll
  for (int j = 0; j < 8; ++j) {
    int c0 = j * 64 + lane * 2;                 // columns [c0, c0+1]
    v2f v = {0.f, 0.f};
    if ((code >> 1) == (c0 >> 1)) v[code & 1] = 1.0f;
    __builtin_nontemporal_store(v, (v2f*)(p + c0));
  }
}

// ------------------------------------------------------------ finalize ----
__global__ __launch_bounds__(512)
void vq_finalize(const int* __restrict__ counts, const float* __restrict__ ws_sse,
                 float* __restrict__ out_loss, float* __restrict__ out_perp) {
  __shared__ float red[KCODES];
  const int t = threadIdx.x;                    // 512 threads
  float s = 0.f;
  for (int i = t; i < NBLK1; i += KCODES) s += ws_sse[i];
  red[t] = s;
  __syncthreads();
  for (int k = 256; k > 0; k >>= 1) { if (t < k) red[t] += red[t + k]; __syncthreads(); }
  float sse_tot = red[0];
  __syncthreads();

  float p = (float)counts[t] / (float)N_TOK;
  red[t] = -p * logf(p + 1e-10f);
  __syncthreads();
  for (int k = 256; k > 0; k >>= 1) { if (t < k) red[t] += red[t + k]; __syncthreads(); }
  if (t == 0) {
    // loss = q_latent + 0.25*e_latent; forward values identical => 1.25 * MSE
    *out_loss = 1.25f * sse_tot / ((float)N_TOK * (float)DIM);
    *out_perp = expf(red[0]);
  }
}

// --------------------------------------------------------------- launch ---
extern "C" void kernel_launch(void* const* d_in, const int* in_sizes, int n_in,
                              void* d_out, int out_size, void* d_ws, size_t ws_size,
                              hipStream_t stream) {
  const float* in  = (const float*)d_in[0];   // [16,64,128,128] f32 NCHW
  const float* emb = (const float*)d_in[1];   // [512,64] f32
  float* out = (float*)d_out;

  char* ws = (char*)d_ws;
  int*   ws_idx = (int*)ws;                                   // N_TOK ints
  int*   counts = (int*)(ws + (size_t)N_TOK * 4);             // 512 ints
  float* ws_sse = (float*)(ws + (size_t)N_TOK * 4 + KCODES * 4); // NBLK1 floats

  float* out_q   = out + OFF_Q;
  float* out_enc = out + OFF_ENC;

  size_t smem = (size_t)(KCODES * EROW + KCODES + 256) * sizeof(float)
              + (size_t)TOK_PER_BLK * sizeof(int);            // ~139.5 KB

  vq_init<<<1, KCODES, 0, stream>>>(counts);
  vq_assign<<<NBLK1, 256, smem, stream>>>(in, emb, out_q, ws_idx, counts, ws_sse);
  vq_encode<<<N_TOK / 8, 256, 0, stream>>>(ws_idx, out_enc);
  vq_finalize<<<1, KCODES, 0, stream>>>(counts, ws_sse, out + OFF_LOSS, out + OFF_PERP);
}